// GATLayer_29557964931464
// MI455X (gfx1250) — compile-verified
//
#include <hip/hip_runtime.h>
#include <hip/hip_bf16.h>

// ---------------------------------------------------------------------------
// GATv2 (2 layers) for MI455X / gfx1250, wave32.
//   dense transforms : fp32 WMMA (v_wmma_f32_16x16x4_f32), exact fp32 math
//   edge phase       : 2-pass fused softmax-aggregate, wave-per-edge,
//                      b128 gathers, u32-ordered atomic max, f32 atomic add
// ---------------------------------------------------------------------------

#define N_NODES   50000
#define N_EDGES   800000
#define NE_TOT    (N_EDGES + N_NODES)   // with self loops
#define IN_DIM    128
#define HEADS     4
#define HID       32
#define HC        (HEADS * HID)         // 128
#define NEG_SLOPE 0.2f

typedef float v2f __attribute__((ext_vector_type(2)));
typedef float v8f __attribute__((ext_vector_type(8)));

// ---- ordered-uint encoding so unsigned atomicMax == float max ----------
__device__ __forceinline__ unsigned ord_encode(float f) {
  unsigned u = __float_as_uint(f);
  return (u & 0x80000000u) ? ~u : (u | 0x80000000u);
}
__device__ __forceinline__ float ord_decode(unsigned o) {
  unsigned u = (o & 0x80000000u) ? (o & 0x7FFFFFFFu) : ~o;
  return __uint_as_float(u);
}
__device__ __forceinline__ float lrelu(float x) {
  return x > 0.0f ? x : NEG_SLOPE * x;
}

// ---------------------------------------------------------------------------
// GEMM: C[M x 128] = A[M x 128] * B[128 x 128], all row-major fp32.
// One wave per 16-row panel; 8 WMMA accumulators cover all 128 columns.
// K loop: 32 steps of v_wmma_f32_16x16x4_f32.
// ---------------------------------------------------------------------------
__global__ void __launch_bounds__(128)
gemm128_wmma(const float* __restrict__ A, const float* __restrict__ B,
             float* __restrict__ C, int M) {
  const int K = 128, NN = 128;
  int wave = threadIdx.x >> 5;
  int tile = blockIdx.x * 4 + wave;          // 16-row panel index
  if (tile * 16 >= M) return;                // wave-uniform: EXEC stays all-1s
  int lane = threadIdx.x & 31;
  int m0   = tile * 16;

  // A fragment (16x4 f32): lanes 0-15 -> K=k,k+1 ; lanes 16-31 -> K=k+2,k+3
  int arow = m0 + (lane & 15);
  int koff = (lane >> 4) * 2;                // 0 or 2
  int ncol = lane & 15;

  v8f acc[8] = {};
  const float* Arow = A + (size_t)arow * K;

  for (int k = 0; k < K; k += 4) {
    v2f a = *(const v2f*)(Arow + k + koff);
#pragma unroll
    for (int n = 0; n < 8; ++n) {
      // B fragment (4x16 f32): VGPR0 row k(+2), VGPR1 row k+1(+3)
      v2f b;
      b.x = B[(size_t)(k + koff)     * NN + n * 16 + ncol];
      b.y = B[(size_t)(k + koff + 1) * NN + n * 16 + ncol];
      acc[n] = __builtin_amdgcn_wmma_f32_16x16x4_f32(
          /*neg_a=*/false, a, /*neg_b=*/false, b,
          /*c_mod=*/(short)0, acc[n], /*reuse_a=*/false, /*reuse_b=*/false);
    }
  }

  // C/D layout: VGPR r -> row m0+r (lanes 0-15) / m0+8+r (lanes 16-31)
  int crow = m0 + (lane >> 4) * 8;
#pragma unroll
  for (int n = 0; n < 8; ++n)
#pragma unroll
    for (int r = 0; r < 8; ++r)
      C[(size_t)(crow + r) * NN + n * 16 + ncol] = acc[n][r];
}

// ---------------------------------------------------------------------------
// Edge pass 1: logits[e][h] = att[h] . leaky_relu(xl[src] + xr[dst])
//              + ordered-uint atomic max into mord[dst][h].
// One wave per edge; lane holds 4 channels (float4); head = lane>>3.
// ---------------------------------------------------------------------------
__global__ void __launch_bounds__(256)
edge_logits(const float* __restrict__ xl, const float* __restrict__ xr,
            const int* __restrict__ src_e, const int* __restrict__ dst_e,
            const float* __restrict__ att,
            float* __restrict__ logits, unsigned* __restrict__ mord) {
  int wave = threadIdx.x >> 5;
  int e = blockIdx.x * 8 + wave;
  if (e >= NE_TOT) return;
  int lane = threadIdx.x & 31;

  int s, d;
  if (e < N_EDGES) { s = src_e[e]; d = dst_e[e]; }
  else             { s = e - N_EDGES; d = s; }   // self loop

  float4 xs = ((const float4*)(xl + (size_t)s * HC))[lane];
  float4 xd = ((const float4*)(xr + (size_t)d * HC))[lane];
  float4 at = ((const float4*)att)[lane];        // att flat [h*HID+c] matches

  float p = at.x * lrelu(xs.x + xd.x)
          + at.y * lrelu(xs.y + xd.y)
          + at.z * lrelu(xs.z + xd.z)
          + at.w * lrelu(xs.w + xd.w);

  // reduce the 8 lanes of each head (lanes [8h .. 8h+7])
  p += __shfl_xor(p, 1, 32);
  p += __shfl_xor(p, 2, 32);
  p += __shfl_xor(p, 4, 32);

  if ((lane & 7) == 0) {
    int h = lane >> 3;
    logits[(size_t)e * HEADS + h] = p;
    atomicMax(&mord[(size_t)d * HEADS + h], ord_encode(p));
  }
}

// ---------------------------------------------------------------------------
// Edge pass 2 (fused): a = exp(logit - max[dst]); denom[dst][h] += a;
//                      acc[dst][h][c] += xl[src][h][c] * a
// (normalization by denom done per-node in finalize -> no 3rd edge pass)
// ---------------------------------------------------------------------------
__global__ void __launch_bounds__(256)
edge_aggregate(const float* __restrict__ xl,
               const int* __restrict__ src_e, const int* __restrict__ dst_e,
               const float* __restrict__ logits, const unsigned* __restrict__ mord,
               float* __restrict__ denom, float* __restrict__ accum) {
  int wave = threadIdx.x >> 5;
  int e = blockIdx.x * 8 + wave;
  if (e >= NE_TOT) return;
  int lane = threadIdx.x & 31;

  int s, d;
  if (e < N_EDGES) { s = src_e[e]; d = dst_e[e]; }
  else             { s = e - N_EDGES; d = s; }

  int h = lane >> 3;
  float m = ord_decode(mord[(size_t)d * HEADS + h]);
  float a = __expf(logits[(size_t)e * HEADS + h] - m);

  if ((lane & 7) == 0)
    atomicAdd(&denom[(size_t)d * HEADS + h], a);

  float4 xs = ((const float4*)(xl + (size_t)s * HC))[lane];
  float* o = accum + (size_t)d * HC + lane * 4;
  atomicAdd(o + 0, xs.x * a);
  atomicAdd(o + 1, xs.y * a);
  atomicAdd(o + 2, xs.z * a);
  atomicAdd(o + 3, xs.w * a);
}

// ---- layer 1 finalize: h1 = relu(acc/denom + b1), concat layout [N,128] ---
__global__ void __launch_bounds__(256)
finalize_concat(const float* __restrict__ accum, const float* __restrict__ denom,
                const float* __restrict__ bias, float* __restrict__ out) {
  int i = blockIdx.x * 256 + threadIdx.x;      // over N_NODES*HC
  if (i >= N_NODES * HC) return;
  int n = i >> 7, hc = i & (HC - 1), h = hc >> 5;
  float v = accum[i] / denom[(size_t)n * HEADS + h] + bias[hc];
  out[i] = v > 0.0f ? v : 0.0f;
}

// ---- layer 2 finalize: h2 = relu(mean_h(acc/denom) + b2), [N,32] ----------
__global__ void __launch_bounds__(256)
finalize_mean(const float* __restrict__ accum, const float* __restrict__ denom,
              const float* __restrict__ bias, float* __restrict__ out) {
  int i = blockIdx.x * 256 + threadIdx.x;      // over N_NODES*HID
  if (i >= N_NODES * HID) return;
  int n = i >> 5, c = i & (HID - 1);
  float sum = 0.0f;
#pragma unroll
  for (int h = 0; h < HEADS; ++h)
    sum += accum[(size_t)n * HC + h * HID + c] / denom[(size_t)n * HEADS + h];
  float v = sum * (1.0f / HEADS) + bias[c];
  out[i] = v > 0.0f ? v : 0.0f;
}

// ---------------------------------------------------------------------------
extern "C" void kernel_launch(void* const* d_in, const int* in_sizes, int n_in,
                              void* d_out, int out_size, void* d_ws, size_t ws_size,
                              hipStream_t stream) {
  const float* x    = (const float*)d_in[0];
  const int*   eix  = (const int*)d_in[1];
  const float* Wl1  = (const float*)d_in[2];
  const float* Wr1  = (const float*)d_in[3];
  const float* att1 = (const float*)d_in[4];
  const float* b1   = (const float*)d_in[5];
  const float* Wl2  = (const float*)d_in[6];
  const float* Wr2  = (const float*)d_in[7];
  const float* att2 = (const float*)d_in[8];
  const float* b2   = (const float*)d_in[9];
  const int* src = eix;
  const int* dst = eix + N_EDGES;

  // ---- workspace carve-up (256B aligned) ----
  char* ws = (char*)d_ws;
  size_t off = 0;
  auto carve = [&](size_t bytes) -> void* {
    void* p = ws + off;
    off += (bytes + 255) & ~(size_t)255;
    return p;
  };
  const size_t NHC = (size_t)N_NODES * HC * sizeof(float);       // 25.6 MB
  float*    xl     = (float*)carve(NHC);
  float*    xr     = (float*)carve(NHC);
  float*    accum  = (float*)carve(NHC);
  float*    h1     = (float*)carve(NHC);
  float*    logits = (float*)carve((size_t)NE_TOT * HEADS * sizeof(float));
  unsigned* mord   = (unsigned*)carve((size_t)N_NODES * HEADS * sizeof(unsigned));
  float*    denom  = (float*)carve((size_t)N_NODES * HEADS * sizeof(float));

  const int GEMM_BLOCKS = (N_NODES / 16 + 3) / 4;   // 3125 tiles / 4 waves
  const int EDGE_BLOCKS = (NE_TOT + 7) / 8;         // 8 waves per block

  // ================= layer 1 =================
  gemm128_wmma<<<GEMM_BLOCKS, 128, 0, stream>>>(x, Wl1, xl, N_NODES);
  gemm128_wmma<<<GEMM_BLOCKS, 128, 0, stream>>>(x, Wr1, xr, N_NODES);
  hipMemsetAsync(mord,  0, (size_t)N_NODES * HEADS * sizeof(unsigned), stream);
  hipMemsetAsync(denom, 0, (size_t)N_NODES * HEADS * sizeof(float),    stream);
  hipMemsetAsync(accum, 0, NHC, stream);
  edge_logits   <<<EDGE_BLOCKS, 256, 0, stream>>>(xl, xr, src, dst, att1, logits, mord);
  edge_aggregate<<<EDGE_BLOCKS, 256, 0, stream>>>(xl, src, dst, logits, mord, denom, accum);
  finalize_concat<<<(N_NODES * HC + 255) / 256, 256, 0, stream>>>(accum, denom, b1, h1);

  // ================= layer 2 =================
  gemm128_wmma<<<GEMM_BLOCKS, 128, 0, stream>>>(h1, Wl2, xl, N_NODES);
  gemm128_wmma<<<GEMM_BLOCKS, 128, 0, stream>>>(h1, Wr2, xr, N_NODES);
  hipMemsetAsync(mord,  0, (size_t)N_NODES * HEADS * sizeof(unsigned), stream);
  hipMemsetAsync(denom, 0, (size_t)N_NODES * HEADS * sizeof(float),    stream);
  hipMemsetAsync(accum, 0, NHC, stream);
  edge_logits   <<<EDGE_BLOCKS, 256, 0, stream>>>(xl, xr, src, dst, att2, logits, mord);
  edge_aggregate<<<EDGE_BLOCKS, 256, 0, stream>>>(xl, src, dst, logits, mord, denom, accum);
  finalize_mean<<<(N_NODES * HID + 255) / 256, 256, 0, stream>>>(accum, denom, b2, (float*)d_out);
}